// CenterDiversityLoss_17497696764047
// MI455X (gfx1250) — compile-verified
//
#include <hip/hip_runtime.h>
#include <hip/hip_bf16.h>

// CDNA5 WMMA vector types: f32 16x16x4 -> A/B are 2 VGPRs (64 f32 / 32 lanes),
// C/D are 8 VGPRs (256 f32 / 32 lanes).
typedef __attribute__((ext_vector_type(2))) float v2f;
typedef __attribute__((ext_vector_type(8))) float v8f;

#define FEAT 128  // feature dim of the reference (centers: N x 128)

// Zero the reduction scratch: ws[0..127] = column sums m, ws[128] = sum of squares S.
__global__ void cdl_init_ws(float* __restrict__ ws) {
    int t = threadIdx.x;
    if (t < FEAT + 1) ws[t] = 0.0f;
}

// Streaming pass: column sums via v_wmma_f32_16x16x4_f32 (A = ones), sum of
// squares via per-lane FMA on the same loaded data. One wave per 16-column
// slice; 8 waves per 256-thread block cover all 128 features.
__global__ __launch_bounds__(256)
void cdl_colsum_sq(const float* __restrict__ centers, float* __restrict__ ws, int n) {
    const int lane    = threadIdx.x & 31;
    const int wave    = threadIdx.x >> 5;   // 0..7
    const int h       = lane >> 4;          // half-wave: 0/1
    const int c       = lane & 15;          // column within slice
    const int colbase = wave * 16;
    const int ntiles  = n >> 4;             // 16-row tiles (n is a multiple of 16)

    v8f acc = {};                            // WMMA accumulator: column sums (rows replicated)
    v2f ones; ones[0] = 1.0f; ones[1] = 1.0f; // A-matrix = all ones (16x4)
    float s = 0.0f;                          // per-lane sum of squares

    for (int t = blockIdx.x; t < ntiles; t += gridDim.x) {
        const float* base = centers + (size_t)t * 16 * FEAT + colbase + c;

        // Prefetch next tile for this wave (global_prefetch_b8).
        if (t + (int)gridDim.x < ntiles) {
            __builtin_prefetch(centers + (size_t)(t + gridDim.x) * 16 * FEAT + colbase + c, 0, 1);
        }

        // Lane (h, c) gathers 8 of the 16 rows of this 16x16 tile; the other
        // half-wave covers the rest. With A = ones, D[m][n] += sum_k B[k][n],
        // so any row->K-slot assignment yields correct column sums.
        v2f b0, b1, b2, b3;
        b0[0] = base[(0  + h) * FEAT];
        b0[1] = base[(2  + h) * FEAT];
        b1[0] = base[(4  + h) * FEAT];
        b1[1] = base[(6  + h) * FEAT];
        b2[0] = base[(8  + h) * FEAT];
        b2[1] = base[(10 + h) * FEAT];
        b3[0] = base[(12 + h) * FEAT];
        b3[1] = base[(14 + h) * FEAT];

        // Sum of squares rides along on the VALU (co-executes with WMMA).
        s = fmaf(b0[0], b0[0], s);
        s = fmaf(b0[1], b0[1], s);
        s = fmaf(b1[0], b1[0], s);
        s = fmaf(b1[1], b1[1], s);
        s = fmaf(b2[0], b2[0], s);
        s = fmaf(b2[1], b2[1], s);
        s = fmaf(b3[0], b3[0], s);
        s = fmaf(b3[1], b3[1], s);

        // 4 chained WMMAs accumulate the 16 rows of this tile into acc.
        acc = __builtin_amdgcn_wmma_f32_16x16x4_f32(false, ones, false, b0, (short)0, acc, false, false);
        acc = __builtin_amdgcn_wmma_f32_16x16x4_f32(false, ones, false, b1, (short)0, acc, false, false);
        acc = __builtin_amdgcn_wmma_f32_16x16x4_f32(false, ones, false, b2, (short)0, acc, false, false);
        acc = __builtin_amdgcn_wmma_f32_16x16x4_f32(false, ones, false, b3, (short)0, acc, false, false);
    }

    // Reduce per-lane sum of squares across the wave (wave32).
    for (int off = 16; off > 0; off >>= 1)
        s += __shfl_down(s, off, 32);
    if (lane == 0)
        unsafeAtomicAdd(&ws[FEAT], s);

    // acc[0]: lanes 0-15 hold D row M=0 (= column sums) for columns colbase+lane;
    // lanes 16-31 hold the identical replicated values (row M=8) -> mask them off.
    if (lane < 16)
        unsafeAtomicAdd(&ws[colbase + lane], acc[0]);
}

// finalize: result = (N*S - ||m||^2) / (count * D)
__global__ void cdl_finalize(const float* __restrict__ ws, float* __restrict__ out, int n) {
    __shared__ float red[FEAT];
    int t = threadIdx.x;
    float mv = ws[t];
    red[t] = mv * mv;
    __syncthreads();
    for (int off = FEAT / 2; off > 0; off >>= 1) {
        if (t < off) red[t] += red[t + off];
        __syncthreads();
    }
    if (t == 0) {
        double S     = (double)ws[FEAT];
        double M2    = (double)red[0];
        double total = (double)n * S - M2;
        double cnt   = (double)n * (double)(n - 1) * 0.5;
        out[0] = (float)(total / (cnt * (double)FEAT));
    }
}

extern "C" void kernel_launch(void* const* d_in, const int* in_sizes, int n_in,
                              void* d_out, int out_size, void* d_ws, size_t ws_size,
                              hipStream_t stream) {
    const float* centers = (const float*)d_in[0];
    float* ws  = (float*)d_ws;
    float* out = (float*)d_out;
    const int n = in_sizes[0] / FEAT;       // 16384 rows
    const int ntiles = n / 16;              // 1024 row-tiles

    hipLaunchKernelGGL(cdl_init_ws, dim3(1), dim3(256), 0, stream, ws);

    int blocks = ntiles < 256 ? ntiles : 256;   // each block: 8 waves x its tile share
    hipLaunchKernelGGL(cdl_colsum_sq, dim3(blocks), dim3(256), 0, stream, centers, ws, n);

    hipLaunchKernelGGL(cdl_finalize, dim3(1), dim3(FEAT), 0, stream, ws, out, n);
}